// Phi_4578435137543
// MI455X (gfx1250) — compile-verified
//
#include <hip/hip_runtime.h>
#include <hip/hip_bf16.h>

typedef __attribute__((ext_vector_type(2))) float v2f;
typedef __attribute__((ext_vector_type(8))) float v8f;

#define N_WIDTH 128
#define N_NODES 513
#define SLICE (N_WIDTH * N_NODES)   // 65664
#define OUT_TOTAL (3 * SLICE)       // 196992

// Order-4 Lagrange basis on nodes {-1,-0.5,0,0.5,1}: values, d/dxt, d2/dxt2,
// already scaled by 1/delta and 1/delta^2 (delta = 1/256, exact in f32).
__device__ __forceinline__ void lagrange15(float xt, float* value /*[16]*/) {
    const float t[5] = {-1.f, -0.5f, 0.f, 0.5f, 1.f};
    const float delta = 0.5f * 4.f * (1.0f - 0.0f) / (float)(N_NODES - 1); // 1/256
    float inv[5][5];
    float ratio[5][5];
#pragma unroll
    for (int j = 0; j < 5; ++j) {
#pragma unroll
        for (int m = 0; m < 5; ++m) {
            if (j == m) { inv[j][m] = 0.f; ratio[j][m] = 1.f; }
            else {
                float d = t[j] - t[m];
                inv[j][m] = 1.f / d;
                ratio[j][m] = (xt - t[m]) / d;
            }
        }
    }
#pragma unroll
    for (int j = 0; j < 5; ++j) {
        float p = 1.f;
#pragma unroll
        for (int m = 0; m < 5; ++m) p *= ratio[j][m];   // ratio[j][j] == 1
        float dp = 0.f, ddp = 0.f;
#pragma unroll
        for (int i = 0; i < 5; ++i) {
            if (i == j) continue;
            float prod3 = 1.f;
#pragma unroll
            for (int m = 0; m < 5; ++m)
                if (m != i && m != j) prod3 *= ratio[j][m];
            dp += inv[j][i] * prod3;
            float inner = 0.f;
#pragma unroll
            for (int m = 0; m < 5; ++m) {
                if (m == i || m == j) continue;
                float prod4 = 1.f;
#pragma unroll
                for (int n = 0; n < 5; ++n)
                    if (n != i && n != j && n != m) prod4 *= ratio[j][n];
                inner += inv[j][m] * prod4;
            }
            ddp += inv[j][i] * inner;
        }
        value[j]      = p;
        value[5 + j]  = dp / delta;
        value[10 + j] = ddp / (delta * delta);
    }
    value[15] = 0.f;
}

// One wave32. Computes the 15 basis values, then broadcasts them through the
// matrix pipe as a rank-1 outer product ones(16) x value(16) using
// V_WMMA_F32_16X16X4_F32 (f32 A/B/C: exact). A has a single 1.0 in column K=0
// (documented A layout: VGPR0 lanes0-15 = K0); B replicates the value row in
// all 4 K rows so the result is layout-robust: D[r][c] = value[c].
__global__ __launch_bounds__(32) void phi_setup_kernel(const float* __restrict__ x_p,
                                                       float* __restrict__ ws) {
    const float x = x_p[0];
    const float x_shift = (float)(N_NODES - 1) * x;          // 512*x
    float fe = floorf(x_shift * 0.25f);
    fe = fminf(fmaxf(fe, 0.f), 127.f);                       // clip id_el
    const int nodes_l = (int)fe * 4;
    const float xt = (x_shift - (float)(nodes_l + 2)) * 0.5f;

    float value[16];
    lagrange15(xt, value);

    const int lane = threadIdx.x;
    v2f a = { (lane < 16) ? 1.f : 0.f, 0.f };   // A[r][k] = delta_{k,0}
    const float bv = value[lane & 15];
    v2f b = { bv, bv };                          // every K-row of B = value[]
    v8f c = {};
    c = __builtin_amdgcn_wmma_f32_16x16x4_f32(
            /*neg_a=*/false, a, /*neg_b=*/false, b,
            /*c_mod=*/(short)0, c, /*reuse_a=*/false, /*reuse_b=*/false);

    // D layout: VGPR0, lanes0-15 -> (M=0, N=lane): D[0][lane] = value[lane]
    if (lane < 16) ws[lane] = c[0];
    if (lane == 0) ((int*)ws)[16] = nodes_l;
}

// Bandwidth pass: out[buf][row][col] = input_buf[sample][row][col], except the
// 5 columns [nodes_l, nodes_l+4] which get the WMMA-broadcast basis values.
__global__ void phi_fill_kernel(const float* __restrict__ b0,
                                const float* __restrict__ b1,
                                const float* __restrict__ b2,
                                const int* __restrict__ sample_p,
                                const float* __restrict__ ws,
                                float* __restrict__ out) {
    const int idx = blockIdx.x * blockDim.x + threadIdx.x;
    if (idx >= OUT_TOTAL) return;
    const int buf = idx / SLICE;
    const int rem = idx - buf * SLICE;
    const int col = rem % N_NODES;

    const int sample = sample_p[0];
    const float* src = (buf == 0) ? b0 : (buf == 1) ? b1 : b2;
    float val = src[(size_t)sample * SLICE + rem];

    const int nodes_l = ((const int*)ws)[16];
    const int d = col - nodes_l;
    if (d >= 0 && d <= 4) val = ws[buf * 5 + d];

    out[idx] = val;
}

extern "C" void kernel_launch(void* const* d_in, const int* in_sizes, int n_in,
                              void* d_out, int out_size, void* d_ws, size_t ws_size,
                              hipStream_t stream) {
    (void)in_sizes; (void)n_in; (void)out_size; (void)ws_size;
    const float* x   = (const float*)d_in[0];
    const float* b0  = (const float*)d_in[1];  // phi_ikp_inner
    const float* b1  = (const float*)d_in[2];  // dphi_ikp_inner
    const float* b2  = (const float*)d_in[3];  // ddphi_ikp_inner
    const int* sample = (const int*)d_in[4];
    float* out = (float*)d_out;
    float* ws  = (float*)d_ws;

    phi_setup_kernel<<<1, 32, 0, stream>>>(x, ws);
    phi_fill_kernel<<<(OUT_TOTAL + 255) / 256, 256, 0, stream>>>(b0, b1, b2, sample, ws, out);
}